// groupingLayer_81243601371766
// MI455X (gfx1250) — compile-verified
//
#include <hip/hip_runtime.h>

#define BATCH   8
#define NDATA   8192
#define NPOINT  1024
#define NSAMPLE 32
#define CFEAT   64
#define R2      0.04f      // RADIUS^2, compare squared distances

#define QTILE   16         // queries per workgroup (one WMMA M-tile)
#define PTILE   256        // points per streamed tile (16 WMMA N-tiles)

#define NP_ELEMS  (BATCH*NPOINT*NSAMPLE*(3+CFEAT))   // new_points floats
#define IDX_ELEMS (BATCH*NPOINT*NSAMPLE)             // idx int32

typedef __attribute__((ext_vector_type(2))) float v2f;
typedef __attribute__((ext_vector_type(8))) float v8f;
typedef __attribute__((ext_vector_type(2))) int   v2i;

// Async global->LDS staging path (gfx1250): guarded, with cooperative-load
// fallback so the kernel always compiles.
#if __has_builtin(__builtin_amdgcn_global_load_async_to_lds_b64) && \
    __has_builtin(__builtin_amdgcn_s_wait_asynccnt)
#define USE_ASYNC_LDS 1
typedef __attribute__((address_space(1))) v2i as1_v2i;   // global-AS int2
typedef __attribute__((address_space(3))) v2i as3_v2i;   // LDS-AS int2
#else
#define USE_ASYNC_LDS 0
#endif

// ---------------------------------------------------------------------------
// Kernel 1: fused WMMA distance tiles + wave-sorted top-32 ball query.
// Grid: BATCH * (NPOINT/QTILE) workgroups, 128 threads (4 waves).
// Per point tile: async-stage 256 points' xyz into LDS, each wave computes 4
// column tiles (4 x v_wmma_f32_16x16x4_f32) into LDS, then maintains top-32
// (dist,idx) lists for 4 queries with lane l holding the l-th smallest
// (sorted insertion via ballot/shfl).
// ---------------------------------------------------------------------------
__global__ __launch_bounds__(128) void ballquery_topk_kernel(
    const float* __restrict__ new_xyz,   // [B][NPOINT][3]
    const float* __restrict__ xyz,       // [B][NDATA][3]
    int* __restrict__ idx_out)           // [B][NPOINT][NSAMPLE]
{
    __shared__ float spts[PTILE * 3];          // staged point xyz, 3 KB
    __shared__ float sdist[PTILE * QTILE];     // [point][query], 16 KB

    const int lane  = threadIdx.x & 31;
    const int w     = threadIdx.x >> 5;        // wave id 0..3
    const int b     = blockIdx.x >> 6;         // 64 query tiles per batch
    const int qbase = (blockIdx.x & 63) * QTILE;

    const float* xyzb = xyz + (size_t)b * NDATA * 3;

    // ---- A operand (loop invariant): row m = [qx, qy, qz, ||q||^2] -------
    // 32-bit A 16x4 layout: lanes 0-15 hold K=0 (V0), K=1 (V1) for M=lane;
    //                       lanes 16-31 hold K=2 (V0), K=3 (V1) for M=lane-16.
    const int am = lane & 15;
    const float* qp = new_xyz + ((size_t)b * NPOINT + qbase + am) * 3;
    const float qx = qp[0], qy = qp[1], qz = qp[2];
    const float qn = qx*qx + qy*qy + qz*qz;
    v2f av;
    av.x = (lane < 16) ? qx : qz;
    av.y = (lane < 16) ? qy : qn;

    // ---- per-wave top-32 lists for 4 queries (lane l = l-th smallest) ----
    float list_d[4];
    int   list_i[4];
#pragma unroll
    for (int qq = 0; qq < 4; ++qq) { list_d[qq] = 3.4e38f; list_i[qq] = 0x7fffffff; }

    for (int pbase = 0; pbase < NDATA; pbase += PTILE) {
        // prefetch next point tile toward L2 (global_prefetch_b8)
        if (pbase + PTILE < NDATA) {
            __builtin_prefetch(xyzb + (size_t)(pbase + PTILE) * 3 + threadIdx.x * 6, 0, 1);
        }

        // -------- stage 256 points (768 floats) into LDS ------------------
        // thread t owns floats [6t, 6t+6): three 8-byte transfers.
        {
            const float* gsrc = xyzb + (size_t)pbase * 3 + threadIdx.x * 6;
            float*       ldst = spts + threadIdx.x * 6;
#if USE_ASYNC_LDS
#pragma unroll
            for (int k = 0; k < 3; ++k) {
                __builtin_amdgcn_global_load_async_to_lds_b64(
                    (as1_v2i*)(gsrc + 2 * k), (as3_v2i*)(ldst + 2 * k), 0, 0);
            }
            __builtin_amdgcn_s_wait_asynccnt(0);
#else
#pragma unroll
            for (int k = 0; k < 3; ++k) {
                *(float2*)(ldst + 2 * k) = *(const float2*)(gsrc + 2 * k);
            }
#endif
        }
        __syncthreads();

        // ---------------- compute phase: 4 WMMAs per wave ----------------
#pragma unroll
        for (int j = 0; j < 4; ++j) {
            const int ct = w * 4 + j;                 // column tile 0..15
            const int n  = lane & 15;                 // column within tile
            const int c  = ct * 16 + n;               // local point 0..255
            const float px = spts[c*3 + 0];
            const float py = spts[c*3 + 1];
            const float pz = spts[c*3 + 2];
            const float pn2 = px*px + py*py + pz*pz;

            // B col n = [-2px, -2py, -2pz, 1]; mirrored 32-bit layout.
            v2f bv;
            bv.x = (lane < 16) ? (-2.0f * px) : (-2.0f * pz);
            bv.y = (lane < 16) ? (-2.0f * py) : 1.0f;

            // C[m][n] = ||p_n||^2 : broadcast over the 8 row-VGPRs.
            v8f cv = {pn2, pn2, pn2, pn2, pn2, pn2, pn2, pn2};

            // D = A*B + C = squared distances, 16 queries x 16 points.
            v8f dv = __builtin_amdgcn_wmma_f32_16x16x4_f32(
                false, av, false, bv, (short)0, cv, false, false);

            // D layout: VGPR k -> M = k (+8 for upper half), N = lane&15.
            const int base = c * QTILE + ((lane < 16) ? 0 : 8);
            *(float4*)(&sdist[base])     = make_float4(dv[0], dv[1], dv[2], dv[3]);
            *(float4*)(&sdist[base + 4]) = make_float4(dv[4], dv[5], dv[6], dv[7]);
        }
        __syncthreads();

        // ---------------- selection phase: 4 queries per wave -------------
#pragma unroll
        for (int qq = 0; qq < 4; ++qq) {
            const int q = w * 4 + qq;
            float kd = __shfl(list_d[qq], 31);        // current 32nd-best
            for (int i = 0; i < PTILE / 32; ++i) {
                const int   c  = i * 32 + lane;
                const float dd = sdist[c * QTILE + q];
                const int   ci = pbase + c;
                unsigned mask = (unsigned)__ballot(dd < kd);
                while (mask) {                         // wave-uniform loop
                    const int src = __ffs((int)mask) - 1;
                    mask &= mask - 1;
                    const float cd  = __shfl(dd, src);
                    const int   cix = __shfl(ci, src);
                    // stable order: smaller dist first, ties by smaller idx
                    const bool less = (list_d[qq] < cd) ||
                                      (list_d[qq] == cd && list_i[qq] < cix);
                    const int pos = __popc((unsigned)__ballot(less));
                    const float pd = __shfl_up(list_d[qq], 1);
                    const int   pi = __shfl_up(list_i[qq], 1);
                    if (lane > pos)       { list_d[qq] = pd; list_i[qq] = pi; }
                    else if (lane == pos) { list_d[qq] = cd; list_i[qq] = cix; }
                }
                kd = __shfl(list_d[qq], 31);
            }
        }
        __syncthreads();
    }

    // ---- radius rule + emit idx ------------------------------------------
#pragma unroll
    for (int qq = 0; qq < 4; ++qq) {
        const int q = w * 4 + qq;
        const int nearest = __shfl(list_i[qq], 0);
        const int out = (list_d[qq] < R2) ? list_i[qq] : nearest;
        idx_out[((size_t)(b * NPOINT + qbase + q)) * NSAMPLE + lane] = out;
    }
}

// ---------------------------------------------------------------------------
// Kernel 2: gather new_points ([xyz | 64 feats]) and grouped_xyz using idx.
// One wave per (b, q, s) sample: coalesced 32-wide feature gathers.
// ---------------------------------------------------------------------------
__global__ __launch_bounds__(256) void gather_kernel(
    const float* __restrict__ xyz,       // [B][NDATA][3]
    const float* __restrict__ points,    // [B][NDATA][CFEAT]
    const int*   __restrict__ idx_in,    // [B][NPOINT][NSAMPLE]
    float* __restrict__ np_out,          // [B][NPOINT][NSAMPLE][67]
    float* __restrict__ gx_out)          // [B][NPOINT][NSAMPLE][3]
{
    const int lane = threadIdx.x & 31;
    const int pair = blockIdx.x * 8 + (threadIdx.x >> 5);   // (b*NPOINT+q)*NSAMPLE+s
    const int b    = pair >> 15;                            // / (NPOINT*NSAMPLE)
    const int pi   = idx_in[pair];

    const float* psrc = points + ((size_t)b * NDATA + pi) * CFEAT;
    float* np = np_out + (size_t)pair * (3 + CFEAT);

    np[3 + lane]      = psrc[lane];
    np[3 + 32 + lane] = psrc[32 + lane];
    if (lane < 3) {
        const float v = xyz[((size_t)b * NDATA + pi) * 3 + lane];
        np[lane] = v;
        gx_out[(size_t)pair * 3 + lane] = v;
    }
}

// ---------------------------------------------------------------------------
extern "C" void kernel_launch(void* const* d_in, const int* in_sizes, int n_in,
                              void* d_out, int out_size, void* d_ws, size_t ws_size,
                              hipStream_t stream) {
    (void)in_sizes; (void)n_in; (void)out_size; (void)d_ws; (void)ws_size;

    const float* new_xyz = (const float*)d_in[0];
    const float* xyz     = (const float*)d_in[1];
    const float* points  = (const float*)d_in[2];

    float* out    = (float*)d_out;
    float* np_out = out;                                   // new_points
    int*   idx_o  = (int*)(out + NP_ELEMS);                // idx (int32 bits)
    float* gx_out = out + NP_ELEMS + IDX_ELEMS;            // grouped_xyz

    ballquery_topk_kernel<<<dim3(BATCH * (NPOINT / QTILE)), dim3(128), 0, stream>>>(
        new_xyz, xyz, idx_o);

    gather_kernel<<<dim3(BATCH * NPOINT * NSAMPLE / 8), dim3(256), 0, stream>>>(
        xyz, points, idx_o, np_out, gx_out);
}